// LocalAttention3D_77017353551959
// MI455X (gfx1250) — compile-verified
//
#include <hip/hip_runtime.h>

// ---------------------------------------------------------------------------
// LocalAttention3D for MI455X (gfx1250), wave32 + V_WMMA_F32_16X16X4_F32.
//
// Shapes (hardcoded from reference): B=2, C=1, D=96, H=64, W=64, NH=4.
// Heads are scalar affines of shared x => scores_h = alpha_h*Gram + gamma_h*r[key]
// (row-constant terms cancel in the per-64-key softmax), and the output
// collapses to  (sum_h (wv_h/Z_h) exp(...)) @ X^T + 64*sum_h bv_h.
//
// LDS layouts are position-major with even padded strides so each WMMA
// fragment's (K, K+1) pair is one 8B-aligned ds_load_b64, except GEMM2's Ks operand (two scalar loads).
// ---------------------------------------------------------------------------

typedef __attribute__((ext_vector_type(2))) float v2f;
typedef __attribute__((ext_vector_type(8))) float v8f;

#define BD 96           // depth (contraction dim for scores)
#define NN 4096         // H*W positions
#define NHEAD 4
#define QS_STR 98       // even + odd-dword spacing: b64-aligned, conflict-free
#define KS_STR 98
#define PT_STR 66

// --- kernel 1: per-position depth rowsum r[b][n] = sum_d x[b,d,n] ----------
__global__ void la3d_rowsum(const float* __restrict__ x, float* __restrict__ r) {
    int n = blockIdx.x * 256 + threadIdx.x;
    if (n >= 2 * NN) return;
    int b = n >> 12;
    int nn = n & (NN - 1);
    const float* p = x + (size_t)b * BD * NN + nn;
    float s = 0.f;
#pragma unroll 8
    for (int d = 0; d < BD; ++d) s += p[(size_t)d * NN];
    r[n] = s;
}

// --- kernel 2: fused scores -> per-row softmax -> combined P -> P @ K^T ----
__global__ __launch_bounds__(512) void la3d_attn(
    const float* __restrict__ x,
    const float* __restrict__ wq, const float* __restrict__ bq,
    const float* __restrict__ wk, const float* __restrict__ bk,
    const float* __restrict__ wv, const float* __restrict__ bv,
    const float* __restrict__ rsum,
    float* __restrict__ out) {

    __shared__ float Qs[64 * QS_STR];      // Qs[q][d]  (position-major)
    __shared__ float Ks[64 * KS_STR];      // Ks[m][d]  (position-major)
    __shared__ float Pt[64 * PT_STR];      // Pt[q][m]  combined probabilities
    __shared__ float redmax[NHEAD * 4 * 64];
    __shared__ float redsum[NHEAD * 4 * 64];
    __shared__ float gmax[NHEAD * 64];     // combined row max (written redundantly)
    __shared__ float rks[64];

    const int tid   = threadIdx.x;
    const int wave  = tid >> 5;
    const int lane  = tid & 31;
    const int lhalf = lane >> 4;   // 0: lanes 0-15, 1: lanes 16-31
    const int lm    = lane & 15;

    const int qt = blockIdx.x;     // 0..63 query tile
    const int b  = blockIdx.y;     // batch

    // Score-phase tile: wave -> (wr, wc) in 4x4 grid of 16x16 tiles.
    const int wr  = wave >> 2;
    const int wc  = wave & 3;
    const int qbS = wr * 16;
    const int kbS = wc * 16;

    // Output-phase tiles: 24 tiles (4 q-blocks x 6 d-blocks) over 16 waves.
    const int qb1 = wave / 6, db1 = wave % 6;
    const int qb2 = (wave + 16) / 6, db2 = (wave + 16) % 6;   // wave < 8 only

    // Per-head softmax coefficients. Row-constant terms cancel in softmax.
    const float inv_scale = 0.17677669529663687f;  // 1/sqrt(32)
    float alpha[NHEAD], gam[NHEAD], wvv[NHEAD];
#pragma unroll
    for (int h = 0; h < NHEAD; ++h) {
        alpha[h] = wq[h] * wk[h] * inv_scale;
        gam[h]   = bq[h] * wk[h] * inv_scale;
        wvv[h]   = wv[h];
    }

    const float* xb = x + (size_t)b * BD * NN;

    // Stage query panel (float4 global loads, transposed LDS stores):
    // Qs[q][d] = x[b, d, qt*64 + q]
    for (int i = tid; i < BD * 16; i += 512) {        // 1536 float4's
        int d = i >> 4, q4 = (i & 15) << 2;
        float4 v = *(const float4*)(xb + (size_t)d * NN + qt * 64 + q4);
        Qs[(q4 + 0) * QS_STR + d] = v.x;
        Qs[(q4 + 1) * QS_STR + d] = v.y;
        Qs[(q4 + 2) * QS_STR + d] = v.z;
        Qs[(q4 + 3) * QS_STR + d] = v.w;
    }

    v8f o1 = {};   // persistent output accumulator, tile 1
    v8f o2 = {};   // persistent output accumulator, tile 2 (wave < 8)

    for (int l = 0; l < 64; ++l) {
        // ---- stage key panel for key-row l: Ks[m][d] = x[b, d, l*64 + m] ----
        const float* kbp = xb + l * 64;
        for (int i = tid; i < BD * 16; i += 512) {
            int d = i >> 4, m4 = (i & 15) << 2;
            float4 v = *(const float4*)(kbp + (size_t)d * NN + m4);
            Ks[(m4 + 0) * KS_STR + d] = v.x;
            Ks[(m4 + 1) * KS_STR + d] = v.y;
            Ks[(m4 + 2) * KS_STR + d] = v.z;
            Ks[(m4 + 3) * KS_STR + d] = v.w;
        }
        if (tid < 64) rks[tid] = rsum[b * NN + l * 64 + tid];
        if (l + 1 < 64) __builtin_prefetch(xb + (l + 1) * 64 + (tid & 63), 0, 0);
        __syncthreads();

        // ---- scores tile: S = Q K^T over depth, K-step 4, b64 fragments ----
        v8f c = {};
        {
            const float* qrow = &Qs[(qbS + lm) * QS_STR + 2 * lhalf];
            const float* krow = &Ks[(kbS + lm) * KS_STR + 2 * lhalf];
#pragma unroll
            for (int d0 = 0; d0 < BD; d0 += 4) {
                v2f a  = *(const v2f*)(qrow + d0);
                v2f bb = *(const v2f*)(krow + d0);
                c = __builtin_amdgcn_wmma_f32_16x16x4_f32(
                        false, a, false, bb, (short)0, c, false, false);
            }
        }

        const float rk = rks[kbS + lm];   // key rowsum (varies with m -> kept)

        // ---- per-head row-max partials (16-lane butterfly, then LDS) ----
#pragma unroll
        for (int h = 0; h < NHEAD; ++h) {
#pragma unroll
            for (int r = 0; r < 8; ++r) {
                float s = alpha[h] * c[r] + gam[h] * rk;
#pragma unroll
                for (int off = 1; off < 16; off <<= 1)
                    s = fmaxf(s, __shfl_xor(s, off, 32));
                if (lm == 0)
                    redmax[(h * 4 + wc) * 64 + qbS + r + 8 * lhalf] = s;
            }
        }
        __syncthreads();

        // ---- per-head exp-sum partials (+ publish combined row max) ----
#pragma unroll
        for (int h = 0; h < NHEAD; ++h) {
#pragma unroll
            for (int r = 0; r < 8; ++r) {
                const int row = qbS + r + 8 * lhalf;
                const float m = fmaxf(
                    fmaxf(redmax[(h * 4 + 0) * 64 + row], redmax[(h * 4 + 1) * 64 + row]),
                    fmaxf(redmax[(h * 4 + 2) * 64 + row], redmax[(h * 4 + 3) * 64 + row]));
                float e = __expf(alpha[h] * c[r] + gam[h] * rk - m);
#pragma unroll
                for (int off = 1; off < 16; off <<= 1)
                    e += __shfl_xor(e, off, 32);
                if (lm == 0) {
                    redsum[(h * 4 + wc) * 64 + row] = e;
                    gmax[h * 64 + row] = m;      // identical value from all wc
                }
            }
        }
        __syncthreads();

        // ---- combined probabilities P = sum_h (wv_h / Z_h) exp(s_h - m_h) ----
        float P[8];
#pragma unroll
        for (int r = 0; r < 8; ++r) P[r] = 0.f;
#pragma unroll
        for (int h = 0; h < NHEAD; ++h) {
#pragma unroll
            for (int r = 0; r < 8; ++r) {
                const int row = qbS + r + 8 * lhalf;
                const float m = gmax[h * 64 + row];
                const float Z = redsum[(h * 4 + 0) * 64 + row] + redsum[(h * 4 + 1) * 64 + row] +
                                redsum[(h * 4 + 2) * 64 + row] + redsum[(h * 4 + 3) * 64 + row];
                P[r] += (wvv[h] / Z) * __expf(alpha[h] * c[r] + gam[h] * rk - m);
            }
        }
#pragma unroll
        for (int r = 0; r < 8; ++r)
            Pt[(qbS + r + 8 * lhalf) * PT_STR + kbS + lm] = P[r];
        __syncthreads();

        // ---- GEMM2: O[q, d] += P[q, m] * K[m][d]  (K-step 4 over m) ----
        {
            const float* pr1 = &Pt[(qb1 * 16 + lm) * PT_STR + 2 * lhalf];
            const float* kc1 = &Ks[(2 * lhalf) * KS_STR + db1 * 16 + lm];
            const float* pr2 = &Pt[(qb2 * 16 + lm) * PT_STR + 2 * lhalf];
            const float* kc2 = &Ks[(2 * lhalf) * KS_STR + db2 * 16 + lm];
#pragma unroll
            for (int m0 = 0; m0 < 64; m0 += 4) {
                {
                    v2f a = *(const v2f*)(pr1 + m0);
                    v2f bb;
                    bb.x = kc1[m0 * KS_STR];
                    bb.y = kc1[(m0 + 1) * KS_STR];
                    o1 = __builtin_amdgcn_wmma_f32_16x16x4_f32(
                            false, a, false, bb, (short)0, o1, false, false);
                }
                if (wave < 8) {
                    v2f a = *(const v2f*)(pr2 + m0);
                    v2f bb;
                    bb.x = kc2[m0 * KS_STR];
                    bb.y = kc2[(m0 + 1) * KS_STR];
                    o2 = __builtin_amdgcn_wmma_f32_16x16x4_f32(
                            false, a, false, bb, (short)0, o2, false, false);
                }
            }
        }
        __syncthreads();   // protect Ks/Pt before next iteration's staging
    }

    // ---- write out: out[b, d, n] = O + 64 * sum_h bv_h ----
    const float bvsum = 64.f * (bv[0] + bv[1] + bv[2] + bv[3]);
    float* ob = out + (size_t)b * BD * NN;
#pragma unroll
    for (int r = 0; r < 8; ++r) {
        const int d = db1 * 16 + lm;
        const int n = qt * 64 + qb1 * 16 + r + 8 * lhalf;
        ob[(size_t)d * NN + n] = o1[r] + bvsum;
    }
    if (wave < 8) {
#pragma unroll
        for (int r = 0; r < 8; ++r) {
            const int d = db2 * 16 + lm;
            const int n = qt * 64 + qb2 * 16 + r + 8 * lhalf;
            ob[(size_t)d * NN + n] = o2[r] + bvsum;
        }
    }
}

extern "C" void kernel_launch(void* const* d_in, const int* in_sizes, int n_in,
                              void* d_out, int out_size, void* d_ws, size_t ws_size,
                              hipStream_t stream) {
    const float* x  = (const float*)d_in[0];
    const float* wq = (const float*)d_in[1];
    const float* bq = (const float*)d_in[2];
    const float* wk = (const float*)d_in[3];
    const float* bk = (const float*)d_in[4];
    const float* wv = (const float*)d_in[5];
    const float* bv = (const float*)d_in[6];
    float* rsum = (float*)d_ws;          // 2*4096 floats = 32 KB scratch
    float* outp = (float*)d_out;

    la3d_rowsum<<<(2 * NN + 255) / 256, 256, 0, stream>>>(x, rsum);

    dim3 grid(NN / 64, 2);               // 64 query tiles x 2 batches
    la3d_attn<<<grid, 512, 0, stream>>>(x, wq, bq, wk, bk, wv, bv, rsum, outp);
}